// IdentityAwareLabelAssignment_10024453669291
// MI455X (gfx1250) — compile-verified
//
#include <hip/hip_runtime.h>
#include <hip/hip_bf16.h>
#include <stdint.h>

// Problem constants (fixed by the reference).
#define BATCH   8
#define HGT     152
#define WID     272
#define HW      (HGT * WID)        // 41344 anchors per batch
#define NGT     100                // gt boxes per batch
#define GPA     (NGT / 4)          // 25 groups of 4 gt per anchor
#define GROUPS  (HW * GPA)         // 1,033,600 float4 outputs per batch
#define MAXANC  16                 // >= max anchors touched by one block (12)
#define EPSF    1e-8f
#define LOG2E   1.442695040888963f
#define LN2     0.693147180559945f

typedef float __attribute__((ext_vector_type(4))) f32x4;

__device__ __forceinline__ f32x4 vsplat(float s) { f32x4 v = {s, s, s, s}; return v; }
__device__ __forceinline__ f32x4 vmax(f32x4 a, f32x4 b) { return __builtin_elementwise_max(a, b); }
__device__ __forceinline__ f32x4 vmin(f32x4 a, f32x4 b) { return __builtin_elementwise_min(a, b); }
__device__ __forceinline__ f32x4 vrcp(f32x4 a) {
    f32x4 r;
    r.x = __builtin_amdgcn_rcpf(a.x);
    r.y = __builtin_amdgcn_rcpf(a.y);
    r.z = __builtin_amdgcn_rcpf(a.z);
    r.w = __builtin_amdgcn_rcpf(a.w);
    return r;
}

__global__ __launch_bounds__(256) void
cost_main(const float* __restrict__ logits,   // [B*HW]   (NUM_CLS == 1)
          const float* __restrict__ bbox,     // [B*HW*4] pixel-space
          const float* __restrict__ gt,       // [B*NGT*4] normalized
          float* __restrict__ out)            // [B*HW*NGT]
{
    __shared__ f32x4 s_aos[NGT];                           // staged gt boxes (AoS)
    __shared__ f32x4 s_gx1[GPA], s_gy1[GPA], s_gx2[GPA],   // SoA views, one f32x4
                     s_gy2[GPA], s_ag[GPA];                // per 4-gt group
    __shared__ float s_cls[MAXANC];                        // 2.5*cls_cost per anchor

    const int b  = blockIdx.y;
    const int g0 = blockIdx.x * 256;
    const int i0 = g0 / GPA;          // first anchor this block touches
    const int t  = threadIdx.x;

    // Waves 0-3: CDNA5 async tensor path — lanes 0..99 each stage one 16B gt
    // box into LDS via GLOBAL_LOAD_ASYNC_TO_LDS_B128 (tracked by ASYNCcnt).
    if (t < NGT) {
        uint32_t lds_off = (uint32_t)(size_t)(&s_aos[t]);  // low 32b = LDS offset
        const float* src = gt + ((size_t)b * NGT + t) * 4;
        asm volatile("global_load_async_to_lds_b128 %0, %1, off"
                     :: "v"(lds_off), "v"(src) : "memory");
    }

    // Wave 4 (independent of the async data): focal class cost for the <=12
    // anchors this block covers, using raw v_exp_f32 / v_log_f32.
    if (t >= 128 && t < 128 + MAXANC) {
        const int a = i0 + (t - 128);
        if (a < HW) {
            const float x  = logits[(size_t)b * HW + a];
            const float p  = __builtin_amdgcn_rcpf(
                                 1.0f + __builtin_amdgcn_exp2f(-LOG2E * x));
            const float lp = __builtin_amdgcn_logf(p + EPSF) * LN2;         // log(p+eps)
            const float ln = __builtin_amdgcn_logf(1.0f - p + EPSF) * LN2;  // log(1-p+eps)
            const float pos = 0.25f * (1.0f - p) * (1.0f - p) * (-lp);
            const float neg = 0.75f * p * p * (-ln);
            s_cls[t - 128] = 2.5f * (pos - neg);
        }
    }

    // Drain async; each staging lane then transposes ITS OWN box (no barrier
    // needed before reading it — per-wave ASYNCcnt==0 means the LDS write landed).
    asm volatile("s_wait_asynccnt 0" ::: "memory");
    if (t < NGT) {
        const f32x4 gb = s_aos[t];
        ((float*)s_gx1)[t] = gb.x;
        ((float*)s_gy1)[t] = gb.y;
        ((float*)s_gx2)[t] = gb.z;
        ((float*)s_gy2)[t] = gb.w;
        ((float*)s_ag )[t] = (gb.z - gb.x) * (gb.w - gb.y);
    }
    __syncthreads();   // single block-wide barrier: SoA + cls now visible to all

    const int g = g0 + t;
    if (g >= GROUPS) return;

    const int i  = g / GPA;       // anchor index
    const int jg = g - i * GPA;   // 4-gt group index

    const float cbase = s_cls[i - i0];

    // Normalized predicted box (one B128 load, reused for 4 gt).
    const f32x4 pb = ((const f32x4* __restrict__)bbox)[(size_t)b * HW + i];
    const float px1 = pb.x * (1.0f / 1088.0f), py1 = pb.y * (1.0f / 608.0f);
    const float px2 = pb.z * (1.0f / 1088.0f), py2 = pb.w * (1.0f / 608.0f);
    const float ap  = (px2 - px1) * (py2 - py1);

    // 4 gt boxes as SoA vectors (one DS_LOAD_B128 each).
    const f32x4 gx1 = s_gx1[jg], gy1 = s_gy1[jg];
    const f32x4 gx2 = s_gx2[jg], gy2 = s_gy2[jg];
    const f32x4 ag  = s_ag[jg];

    // Intersection / enclosure corners (min/max shared with the L1 cost).
    const f32x4 ltx = vmax(gx1, vsplat(px1)), lty = vmax(gy1, vsplat(py1));
    const f32x4 rbx = vmin(gx2, vsplat(px2)), rby = vmin(gy2, vsplat(py2));
    const f32x4 ex  = vmin(gx1, vsplat(px1)), ey  = vmin(gy1, vsplat(py1));
    const f32x4 ex2 = vmax(gx2, vsplat(px2)), ey2 = vmax(gy2, vsplat(py2));

    // |a-b| = max(a,b) - min(a,b)  =>  L1 comes free from the GIoU corners.
    const f32x4 l1 = ((ltx + lty) + (ex2 + ey2)) - ((ex + ey) + (rbx + rby));

    const f32x4 zero  = vsplat(0.0f);
    const f32x4 iw    = vmax(rbx - ltx, zero);
    const f32x4 ih    = vmax(rby - lty, zero);
    const f32x4 inter = iw * ih;
    const f32x4 uni   = (ag + ap) - inter;
    const f32x4 iou   = inter * vrcp(uni + EPSF);
    const f32x4 enc   = (ex2 - ex) * (ey2 - ey);    // >= 0 by construction
    const f32x4 q     = (enc - uni) * vrcp(enc + EPSF);

    // C = 2.5*cls + 5*l1 - 2*giou,  giou = iou - q
    const f32x4 res = ((cbase + 5.0f * l1) + 2.0f * q) - 2.0f * iou;

    // Streaming NT B128 store; float4 offset = b*GROUPS + g -> fully coalesced.
    f32x4* __restrict__ op = ((f32x4* __restrict__)out) + (size_t)b * GROUPS + g;
    __builtin_nontemporal_store(res, op);
}

extern "C" void kernel_launch(void* const* d_in, const int* in_sizes, int n_in,
                              void* d_out, int out_size, void* d_ws, size_t ws_size,
                              hipStream_t stream) {
    (void)in_sizes; (void)n_in; (void)out_size; (void)d_ws; (void)ws_size;
    const float* logits = (const float*)d_in[0];   // cls_pred_logits (B,H,W,1)
    const float* bbox   = (const float*)d_in[1];   // bbox_preds     (B,HW,4)
    const float* gt     = (const float*)d_in[2];   // gt_boxes       (B,NGT,4)
    // d_in[3] = gt_cls: NUM_CLS==1 -> class index is provably 0; unused.
    float* out = (float*)d_out;                    // C (B,HW,NGT)

    const dim3 grid((GROUPS + 255) / 256, BATCH, 1);
    cost_main<<<grid, 256, 0, stream>>>(logits, bbox, gt, out);
}